// Arma_8967891714116
// MI455X (gfx1250) — compile-verified
//
#include <hip/hip_runtime.h>
#include <hip/hip_bf16.h>

typedef float v2f __attribute__((ext_vector_type(2)));
typedef float v8f __attribute__((ext_vector_type(8)));

#define NND 100000   // nodes
#define FIN 512      // input features
#define HID 16       // hidden
#define NCL 40       // classes
// K stacks = 2

// ---------------------------------------------------------------------------
// GCN norm: degree over dst, dinv = rsqrt(deg), ew = dinv[src]*dinv[dst]
// ---------------------------------------------------------------------------
__global__ void deg_kernel(const int* __restrict__ dst, float* __restrict__ deg, int E) {
    int i = blockIdx.x * blockDim.x + threadIdx.x;
    if (i < E) atomicAdd(&deg[dst[i]], 1.0f);
}

__global__ void dinv_kernel(float* __restrict__ deg, int n) {
    int i = blockIdx.x * blockDim.x + threadIdx.x;
    if (i < n) {
        float d = deg[i];
        deg[i] = (d > 0.0f) ? rsqrtf(d) : 0.0f;
    }
}

__global__ void ew_kernel(const int* __restrict__ src, const int* __restrict__ dst,
                          const float* __restrict__ dinv, float* __restrict__ ew, int E) {
    int i = blockIdx.x * blockDim.x + threadIdx.x;
    if (i < E) ew[i] = dinv[src[i]] * dinv[dst[i]];
}

// ---------------------------------------------------------------------------
// Layer-1 GEMM via V_WMMA_F32_16X16X4_F32.
// Block = 128 threads = 4 waves; wave w handles matrix:
//   w0: k=0 iw1 -> hmsg1[0], w1: k=0 rw1(+b1) -> pre1[0],
//   w2: k=1 iw1 -> hmsg1[1], w3: k=1 rw1(+b1) -> pre1[1].
// A (16x4 f32): lanes 0-15 hold rows M=0..15, v0=K+0,v1=K+1; lanes 16-31 K+2,K+3.
// B (4x16 f32): v0: lanes0-15 row K+0 / lanes16-31 row K+2; v1: K+1 / K+3.
// C/D (16x16 f32, 8 VGPRs): vgpr i -> row i (lanes 0-15) / row i+8 (lanes 16-31).
// ---------------------------------------------------------------------------
__global__ __launch_bounds__(128) void gemm1_kernel(
    const float* __restrict__ x,     // [NND, FIN]
    const float* __restrict__ iw1,   // [2, FIN, HID]
    const float* __restrict__ rw1,   // [2, FIN, HID]
    const float* __restrict__ b1,    // [2, 1, HID]
    float* __restrict__ hmsg1,       // [2, NND, HID]
    float* __restrict__ pre1)        // [2, NND, HID]
{
    const int rb   = blockIdx.x;            // 16-row block of x
    const int wave = threadIdx.x >> 5;      // 0..3
    const int lane = threadIdx.x & 31;
    const int k    = wave >> 1;             // ARMA stack
    const bool isRoot = (wave & 1) != 0;

    const float* __restrict__ W = (isRoot ? rw1 : iw1) + (size_t)k * FIN * HID;

    const int mrow = rb * 16 + (lane & 15);
    const int koff = (lane >> 4) * 2;
    const float* __restrict__ xr = x + (size_t)mrow * FIN;

    v8f c = {0.f, 0.f, 0.f, 0.f, 0.f, 0.f, 0.f, 0.f};
    const int bcol = lane & 15;

    for (int f0 = 0; f0 < FIN; f0 += 4) {
        v2f a, b;
        a.x = xr[f0 + koff];
        a.y = xr[f0 + koff + 1];
        const float* wr = W + (size_t)(f0 + koff) * HID + bcol;
        b.x = wr[0];
        b.y = wr[HID];
        c = __builtin_amdgcn_wmma_f32_16x16x4_f32(
                false, a, false, b, (short)0, c, false, false);
    }

    const float bias = isRoot ? b1[k * HID + bcol] : 0.0f;
    float* __restrict__ dp = (isRoot ? pre1 : hmsg1) + (size_t)k * NND * HID;
    const int rhalf = (lane >> 4) << 3;
#pragma unroll
    for (int i = 0; i < 8; ++i) {
        int r = rb * 16 + i + rhalf;
        dp[(size_t)r * HID + bcol] = c[i] + bias;
    }
}

// ---------------------------------------------------------------------------
// Scatter 1: 32 threads per edge; lanes 0-15 -> k=0, lanes 16-31 -> k=1,
// each lane one of the 16 features (coalesced gather + L2-resident atomics).
// ---------------------------------------------------------------------------
__global__ __launch_bounds__(256) void scatter1_kernel(
    const int* __restrict__ src, const int* __restrict__ dst,
    const float* __restrict__ ew,
    const float* __restrict__ hmsg1,   // [2, NND, HID]
    float* __restrict__ pre1,          // [2, NND, HID]
    int E)
{
    int gid = blockIdx.x * 256 + threadIdx.x;
    if (gid >= E * 32) return;
    int e    = gid >> 5;
    int lane = gid & 31;
    int k    = lane >> 4;
    int o    = lane & 15;
    int s = src[e], d = dst[e];
    float w = ew[e];
    float v = hmsg1[((size_t)k * NND + s) * HID + o];
    atomicAdd(&pre1[((size_t)k * NND + d) * HID + o], w * v);
}

// h2 = mean_k relu(pre1[k])
__global__ void finish1_kernel(const float* __restrict__ pre1, float* __restrict__ h2) {
    int i = blockIdx.x * blockDim.x + threadIdx.x;
    if (i < NND * HID) {
        float a = pre1[i];
        float b = pre1[(size_t)NND * HID + i];
        h2[i] = 0.5f * (fmaxf(a, 0.0f) + fmaxf(b, 0.0f));
    }
}

// ---------------------------------------------------------------------------
// Layer-2 GEMM: h2 [NND,16] x W2 [16,40] via WMMA; 3 column tiles cover 40.
// ---------------------------------------------------------------------------
__global__ __launch_bounds__(128) void gemm2_kernel(
    const float* __restrict__ h2,    // [NND, HID]
    const float* __restrict__ iw2,   // [2, HID, NCL]
    const float* __restrict__ rw2,   // [2, HID, NCL]
    const float* __restrict__ b2,    // [2, 1, NCL]
    float* __restrict__ hmsg2,       // [2, NND, NCL]
    float* __restrict__ pre2)        // [2, NND, NCL]
{
    const int rb   = blockIdx.x;
    const int wave = threadIdx.x >> 5;
    const int lane = threadIdx.x & 31;
    const int k    = wave >> 1;
    const bool isRoot = (wave & 1) != 0;

    const float* __restrict__ W = (isRoot ? rw2 : iw2) + (size_t)k * HID * NCL;
    const int mrow = rb * 16 + (lane & 15);
    const int koff = (lane >> 4) * 2;
    const float* __restrict__ hr = h2 + (size_t)mrow * HID;

    // A operands are reused across the 3 column tiles: preload 8 scalars.
    float a0[4], a1[4];
#pragma unroll
    for (int t = 0; t < 4; ++t) {
        a0[t] = hr[4 * t + koff];
        a1[t] = hr[4 * t + koff + 1];
    }

    const int bcol16 = lane & 15;
    const int rhalf  = (lane >> 4) << 3;
    float* __restrict__ dp = (isRoot ? pre2 : hmsg2) + (size_t)k * NND * NCL;

    for (int ct = 0; ct < 3; ++ct) {
        const int col = ct * 16 + bcol16;
        const bool colok = (col < NCL);
        v8f c = {0.f, 0.f, 0.f, 0.f, 0.f, 0.f, 0.f, 0.f};
#pragma unroll
        for (int t = 0; t < 4; ++t) {
            const int f0 = 4 * t;
            v2f a, b;
            a.x = a0[t];
            a.y = a1[t];
            b.x = colok ? W[(size_t)(f0 + koff) * NCL + col] : 0.0f;
            b.y = colok ? W[(size_t)(f0 + koff + 1) * NCL + col] : 0.0f;
            c = __builtin_amdgcn_wmma_f32_16x16x4_f32(
                    false, a, false, b, (short)0, c, false, false);
        }
        if (colok) {
            const float bias = isRoot ? b2[k * NCL + col] : 0.0f;
#pragma unroll
            for (int i = 0; i < 8; ++i) {
                int r = rb * 16 + i + rhalf;
                dp[(size_t)r * NCL + col] = c[i] + bias;
            }
        }
    }
}

// Scatter 2: one thread per (edge, stack, feature) — E*80 threads.
__global__ __launch_bounds__(256) void scatter2_kernel(
    const int* __restrict__ src, const int* __restrict__ dst,
    const float* __restrict__ ew,
    const float* __restrict__ hmsg2,   // [2, NND, NCL]
    float* __restrict__ pre2,          // [2, NND, NCL]
    int E)
{
    int gid = blockIdx.x * 256 + threadIdx.x;
    if (gid >= E * 80) return;
    int o    = gid % NCL;
    int rest = gid / NCL;
    int k    = rest & 1;
    int e    = rest >> 1;
    int s = src[e], d = dst[e];
    float w = ew[e];
    float v = hmsg2[((size_t)k * NND + s) * NCL + o];
    atomicAdd(&pre2[((size_t)k * NND + d) * NCL + o], w * v);
}

// out = mean_k relu(pre2[k])
__global__ void finish2_kernel(const float* __restrict__ pre2, float* __restrict__ out) {
    int i = blockIdx.x * blockDim.x + threadIdx.x;
    if (i < NND * NCL) {
        float a = pre2[i];
        float b = pre2[(size_t)NND * NCL + i];
        out[i] = 0.5f * (fmaxf(a, 0.0f) + fmaxf(b, 0.0f));
    }
}

// ---------------------------------------------------------------------------
extern "C" void kernel_launch(void* const* d_in, const int* in_sizes, int n_in,
                              void* d_out, int out_size, void* d_ws, size_t ws_size,
                              hipStream_t stream) {
    const float* x   = (const float*)d_in[0];
    const int* eidx  = (const int*)d_in[1];
    const float* iw1 = (const float*)d_in[2];
    const float* rw1 = (const float*)d_in[3];
    const float* b1  = (const float*)d_in[4];
    const float* iw2 = (const float*)d_in[5];
    const float* rw2 = (const float*)d_in[6];
    const float* b2  = (const float*)d_in[7];

    const int E = in_sizes[1] / 2;
    const int* src = eidx;
    const int* dst = eidx + E;

    // Workspace layout (floats)
    float* ws = (float*)d_ws;
    size_t off = 0;
    float* deg   = ws + off; off += NND;                       // also holds dinv
    float* ew    = ws + off; off += (size_t)E;
    float* hmsg1 = ws + off; off += (size_t)2 * NND * HID;
    float* pre1  = ws + off; off += (size_t)2 * NND * HID;
    float* h2    = ws + off; off += (size_t)NND * HID;
    float* hmsg2 = ws + off; off += (size_t)2 * NND * NCL;
    float* pre2  = ws + off; off += (size_t)2 * NND * NCL;

    float* out = (float*)d_out;

    // --- GCN norm ---
    hipMemsetAsync(deg, 0, (size_t)NND * sizeof(float), stream);
    deg_kernel<<<(E + 255) / 256, 256, 0, stream>>>(dst, deg, E);
    dinv_kernel<<<(NND + 255) / 256, 256, 0, stream>>>(deg, NND);
    ew_kernel<<<(E + 255) / 256, 256, 0, stream>>>(src, dst, deg, ew, E);

    // --- Layer 1 ---
    gemm1_kernel<<<NND / 16, 128, 0, stream>>>(x, iw1, rw1, b1, hmsg1, pre1);
    {
        long long total = (long long)E * 32;
        int blocks = (int)((total + 255) / 256);
        scatter1_kernel<<<blocks, 256, 0, stream>>>(src, dst, ew, hmsg1, pre1, E);
    }
    finish1_kernel<<<(NND * HID + 255) / 256, 256, 0, stream>>>(pre1, h2);

    // --- Layer 2 ---
    gemm2_kernel<<<NND / 16, 128, 0, stream>>>(h2, iw2, rw2, b2, hmsg2, pre2);
    {
        long long total = (long long)E * 80;
        int blocks = (int)((total + 255) / 256);
        scatter2_kernel<<<blocks, 256, 0, stream>>>(src, dst, ew, hmsg2, pre2, E);
    }
    finish2_kernel<<<(NND * NCL + 255) / 256, 256, 0, stream>>>(pre2, out);
}